// GCF_3513283248500
// MI455X (gfx1250) — compile-verified
//
#include <hip/hip_runtime.h>
#include <hip/hip_bf16.h>

// ---------------------------------------------------------------------------
// LightGCN-style pipeline for MI455X (gfx1250, wave32).
//   e1 = A @ ego          (COO scatter w/ global_atomic_add_f32, L2-resident)
//   e2 = A @ e1
//   final = (ego + e1 + e2)/3 gathered at batch rows, fused into MLP kernel
//   MLP 200->64->32->1 using V_WMMA_F32_16X16X4_F32 (fp32 tensor path)
// ---------------------------------------------------------------------------

#define N_USERS 100000
#define N_ITEMS 50000
#define N_NODES 150000
#define EMB     100
#define BATCH   16384

typedef __attribute__((ext_vector_type(2))) float v2f;
typedef __attribute__((ext_vector_type(8))) float v8f;

// ---------------------------------------------------------------------------
// Zero-fill workspace (grid-stride). Needed every call: atomics accumulate.
// ---------------------------------------------------------------------------
__global__ void gcf_zero_kernel(float* __restrict__ p, long long n) {
    long long i = (long long)blockIdx.x * blockDim.x + threadIdx.x;
    long long stride = (long long)gridDim.x * blockDim.x;
    for (; i < n; i += stride) p[i] = 0.0f;
}

// ---------------------------------------------------------------------------
// COO SpMM scatter: dst[row] += val * src[col]
// One thread per (edge, 4-dim chunk): 25 chunks cover EMB=100.
// Row pitch is 400 B -> float4 loads are 16B-aligned.
// mode 0: src row comes from user_emb/item_emb split (ego); mode 1: srcFull.
// ---------------------------------------------------------------------------
__global__ void gcf_spmm_kernel(const int* __restrict__ row,
                                const int* __restrict__ col,
                                const float* __restrict__ val,
                                const float* __restrict__ user_emb,
                                const float* __restrict__ item_emb,
                                const float* __restrict__ srcFull,
                                float* __restrict__ dst,
                                int nnz, int mode) {
    long long gid = (long long)blockIdx.x * blockDim.x + threadIdx.x;
    long long total = (long long)nnz * 25;
    if (gid >= total) return;
    int e = (int)(gid / 25);
    int q = (int)(gid % 25);
    int d = q * 4;

    int r = row[e];
    int c = col[e];
    float v = val[e];

    const float* srow;
    if (mode == 0) {
        srow = (c < N_USERS) ? (user_emb + (long long)c * EMB)
                             : (item_emb + (long long)(c - N_USERS) * EMB);
    } else {
        srow = srcFull + (long long)c * EMB;
    }
    float4 x = *(const float4*)(srow + d);

    float* drow = dst + (long long)r * EMB + d;
    atomicAdd(drow + 0, v * x.x);
    atomicAdd(drow + 1, v * x.y);
    atomicAdd(drow + 2, v * x.z);
    atomicAdd(drow + 3, v * x.w);
}

// ---------------------------------------------------------------------------
// Fused gather + 3-layer MLP with fp32 WMMA (16x16x4).
// Block = 128 threads = 4 waves; each wave owns a 16-row batch tile.
// Per-wave LDS buffer: 3200 floats (16x200 X tile; reused for 16x64 H1 and
// 16x32 H2 tiles). 4 waves * 3200 * 4B = 51.2 KB static LDS.
// ---------------------------------------------------------------------------
__global__ __launch_bounds__(128) void gcf_mlp_kernel(
        const float* __restrict__ user_emb,
        const float* __restrict__ item_emb,
        const float* __restrict__ e1,
        const float* __restrict__ e2,
        const int* __restrict__ userIdx,
        const int* __restrict__ itemIdx,
        const float* __restrict__ W1, const float* __restrict__ b1,
        const float* __restrict__ W2, const float* __restrict__ b2,
        const float* __restrict__ W3, const float* __restrict__ b3,
        float* __restrict__ out) {
    __shared__ float smem[4 * 3200];

    const int wave = threadIdx.x >> 5;
    const int lane = threadIdx.x & 31;
    const int half = lane >> 4;     // 0: lanes 0-15, 1: lanes 16-31
    const int l16  = lane & 15;

    float* xs = smem + wave * 3200;
    const int rowBase = blockIdx.x * 64 + wave * 16;   // 16 batch rows per wave

    // ---- stage X tile: 16 rows x 200 cols = (final[u] , final[item]) -------
    for (int t = lane; t < 16 * 200; t += 32) {
        int m = t / 200;
        int d = t % 200;
        int bi = rowBase + m;
        float v;
        if (d < EMB) {
            int idx = userIdx[bi];
            long long o = (long long)idx * EMB + d;
            v = (user_emb[o] + e1[o] + e2[o]) * (1.0f / 3.0f);
        } else {
            int idx = itemIdx[bi];
            int dd = d - EMB;
            long long oe = (long long)idx * EMB + dd;                 // ego (item table)
            long long on = (long long)(idx + N_USERS) * EMB + dd;     // node row in e1/e2
            v = (item_emb[oe] + e1[on] + e2[on]) * (1.0f / 3.0f);
        }
        xs[m * 200 + d] = v;
    }
    __syncthreads();

    // ---- layer 1: X[16x200] @ W1[200x64] -> H1[16x64], relu + b1 -----------
    v8f c1[4] = {v8f{}, v8f{}, v8f{}, v8f{}};
    for (int k0 = 0; k0 < 200; k0 += 4) {
        // A 16x4 fragment: lanes0-15 hold K=k0,k0+1; lanes16-31 hold K=k0+2,k0+3
        v2f a;
        a.x = xs[l16 * 200 + k0 + 2 * half + 0];
        a.y = xs[l16 * 200 + k0 + 2 * half + 1];
#pragma unroll
        for (int n = 0; n < 4; ++n) {
            int colb = n * 16 + l16;
            v2f b;
            b.x = W1[(k0 + 2 * half + 0) * 64 + colb];
            b.y = W1[(k0 + 2 * half + 1) * 64 + colb];
            c1[n] = __builtin_amdgcn_wmma_f32_16x16x4_f32(
                        false, a, false, b, (short)0, c1[n], false, false);
        }
    }
    __syncthreads();

    // ---- bias + relu, spill H1 (16x64) to LDS ------------------------------
#pragma unroll
    for (int n = 0; n < 4; ++n) {
        int colb = n * 16 + l16;
        float bias = b1[colb];
#pragma unroll
        for (int j = 0; j < 8; ++j) {
            int m = j + 8 * half;     // C/D layout: VGPR j -> M = j + 8*half
            float v = c1[n][j] + bias;
            xs[m * 64 + colb] = v > 0.0f ? v : 0.0f;
        }
    }
    __syncthreads();

    // ---- layer 2: H1[16x64] @ W2[64x32] -> H2[16x32] -----------------------
    v8f c2[2] = {v8f{}, v8f{}};
    for (int k0 = 0; k0 < 64; k0 += 4) {
        v2f a;
        a.x = xs[l16 * 64 + k0 + 2 * half + 0];
        a.y = xs[l16 * 64 + k0 + 2 * half + 1];
#pragma unroll
        for (int n = 0; n < 2; ++n) {
            int colb = n * 16 + l16;
            v2f b;
            b.x = W2[(k0 + 2 * half + 0) * 32 + colb];
            b.y = W2[(k0 + 2 * half + 1) * 32 + colb];
            c2[n] = __builtin_amdgcn_wmma_f32_16x16x4_f32(
                        false, a, false, b, (short)0, c2[n], false, false);
        }
    }
    __syncthreads();

    // ---- bias b2, spill H2 (16x32) to LDS ----------------------------------
#pragma unroll
    for (int n = 0; n < 2; ++n) {
        int colb = n * 16 + l16;
        float bias = b2[colb];
#pragma unroll
        for (int j = 0; j < 8; ++j) {
            int m = j + 8 * half;
            xs[m * 32 + colb] = c2[n][j] + bias;
        }
    }
    __syncthreads();

    // ---- layer 3: H2[16x32] @ W3[32x1] + b3 --------------------------------
    if (lane < 16) {
        float s = b3[0];
#pragma unroll
        for (int j = 0; j < 32; ++j) s += xs[lane * 32 + j] * W3[j];
        out[rowBase + lane] = s;
    }
}

// ---------------------------------------------------------------------------
extern "C" void kernel_launch(void* const* d_in, const int* in_sizes, int n_in,
                              void* d_out, int out_size, void* d_ws, size_t ws_size,
                              hipStream_t stream) {
    const float* user_emb = (const float*)d_in[0];
    const float* item_emb = (const float*)d_in[1];
    const int*   adj_row  = (const int*)d_in[2];
    const int*   adj_col  = (const int*)d_in[3];
    const float* adj_val  = (const float*)d_in[4];
    const int*   userIdx  = (const int*)d_in[5];
    const int*   itemIdx  = (const int*)d_in[6];
    const float* W1 = (const float*)d_in[7];
    const float* b1 = (const float*)d_in[8];
    const float* W2 = (const float*)d_in[9];
    const float* b2 = (const float*)d_in[10];
    const float* W3 = (const float*)d_in[11];
    const float* b3 = (const float*)d_in[12];
    float* out = (float*)d_out;

    const int nnz = in_sizes[2];                       // 2,000,000

    // workspace: e1 | e2, each N_NODES*EMB floats (60 MB each)
    float* e1 = (float*)d_ws;
    float* e2 = e1 + (long long)N_NODES * EMB;

    // zero e1,e2 every call (atomics accumulate; harness does not re-poison)
    long long zn = 2LL * N_NODES * EMB;
    gcf_zero_kernel<<<2048, 256, 0, stream>>>(e1, zn);

    // layer 1: e1 = A @ ego (ego read from split user/item tables)
    long long work = (long long)nnz * 25;
    int blocks = (int)((work + 255) / 256);
    gcf_spmm_kernel<<<blocks, 256, 0, stream>>>(adj_row, adj_col, adj_val,
                                                user_emb, item_emb, nullptr,
                                                e1, nnz, 0);
    // layer 2: e2 = A @ e1
    gcf_spmm_kernel<<<blocks, 256, 0, stream>>>(adj_row, adj_col, adj_val,
                                                user_emb, item_emb, e1,
                                                e2, nnz, 1);

    // fused gather + MLP: 16 rows per wave, 4 waves per block
    gcf_mlp_kernel<<<BATCH / 64, 128, 0, stream>>>(user_emb, item_emb, e1, e2,
                                                   userIdx, itemIdx,
                                                   W1, b1, W2, b2, W3, b3, out);
}